// LaplacianPyramid_58128087384510
// MI455X (gfx1250) — compile-verified
//
#include <hip/hip_runtime.h>

// Unaligned-tolerant 2-float pair (4-byte aligned B64 global load is legal on
// gfx1250 in DWORD alignment mode).
struct __attribute__((packed, aligned(4))) F2 {
    float a, b;
};

typedef float vf2 __attribute__((ext_vector_type(2)));

struct LvlCoords {
    int off0, off1;               // element offsets (channel 0) of the two row-pairs
    float w00, w01, w10, w11;     // validity-folded bilinear weights mapped onto the pairs
};

template <int W>
__device__ __forceinline__ LvlCoords make_coords(float gx, float gy) {
    constexpr int H = W;
    // align_corners=False mapping (f32, matches reference)
    const float ix = ((gx + 1.0f) * (float)W - 1.0f) * 0.5f;
    const float iy = ((gy + 1.0f) * (float)H - 1.0f) * 0.5f;
    const float fx0 = floorf(ix);
    const float fy0 = floorf(iy);
    const int x0 = (int)fx0;          // in [-1, W-1] for gx in [-1,1)
    const int y0 = (int)fy0;          // in [-1, H-1]
    const float wx1 = ix - fx0, wx0 = 1.0f - wx1;
    const float wy1 = iy - fy0, wy0 = 1.0f - wy1;
    // zeros-padding: fold validity into weights (value clamping is then harmless)
    const float vx0 = (x0 >= 0)     ? wx0 : 0.0f;   // x0 <= W-1 always
    const float vx1 = (x0 <= W - 2) ? wx1 : 0.0f;   // x1 = x0+1 valid iff x0 <= W-2
    const float vy0 = (y0 >= 0)     ? wy0 : 0.0f;
    const float vy1 = (y0 <= H - 2) ? wy1 : 0.0f;
    // one B64 per row covers both taps: base column clamped to [0, W-2]
    const int xlo = min(max(x0, 0), W - 2);
    const int y0c = max(y0, 0);
    const int y1c = min(y0 + 1, H - 1);
    // remap tap weights onto (pair[0], pair[1]) at columns (xlo, xlo+1):
    //   x0 == xlo      : interior -> (vx0, vx1)
    //   x0 == -1       : xlo = 0   -> pair[0] is tap1 -> (vx1, 0)   [vx0 == 0]
    //   x0 == W-1      : xlo = W-2 -> pair[1] is tap0 -> (0, vx0)   [vx1 == 0]
    float u0 = (x0 == xlo) ? vx0 : 0.0f;
    u0 += (x0 + 1 == xlo) ? vx1 : 0.0f;
    const float u1 = (x0 == xlo) ? vx1 : ((x0 == xlo + 1) ? vx0 : 0.0f);
    LvlCoords r;
    r.off0 = y0c * W + xlo;
    r.off1 = y1c * W + xlo;
    r.w00 = vy0 * u0;
    r.w01 = vy0 * u1;
    r.w10 = vy1 * u0;
    r.w11 = vy1 * u1;
    return r;
}

template <int W, int CH>
__device__ __forceinline__ void accum_level(const float* __restrict__ tex,
                                            const LvlCoords lc, int cbase,
                                            float* __restrict__ acc) {
    constexpr size_t HW = (size_t)W * (size_t)W;
    const float* p0 = tex + (size_t)cbase * HW + (size_t)lc.off0;
    const float* p1 = tex + (size_t)cbase * HW + (size_t)lc.off1;
#pragma unroll
    for (int c = 0; c < CH; ++c) {
        const F2 a = *(const F2*)(p0 + (size_t)c * HW);  // taps (y0,x0),(y0,x1)
        const F2 b = *(const F2*)(p1 + (size_t)c * HW);  // taps (y1,x0),(y1,x1)
        acc[c] += lc.w00 * a.a + lc.w01 * a.b + lc.w10 * b.a + lc.w11 * b.b;
    }
}

// Block: 256 threads walk output pixels (n,ho,wo) -> coalesced grid reads +
// coalesced NT output stores. blockIdx.y selects an 8-channel chunk so each
// thread keeps only 8 accumulators -> small VGPR footprint, high wave32
// occupancy to hide random-gather latency (all textures are L2-resident).
__global__ __launch_bounds__(256) void grid_pyr_58128087384510_kernel(
    const float* __restrict__ x,
    const float* __restrict__ t0, const float* __restrict__ t1,
    const float* __restrict__ t2, const float* __restrict__ t3,
    float* __restrict__ out) {
    const int p = blockIdx.x * 256 + threadIdx.x;  // pixel id in [0, 4*512*512)
    const int cbase = blockIdx.y * 8;              // channel chunk base

    // grid coords: streamed once -> non-temporal load keeps L2 for textures
    const vf2 g = __builtin_nontemporal_load((const vf2*)x + p);
    const float gx = g.x, gy = g.y;

    const LvlCoords c0 = make_coords<1024>(gx, gy);
    const LvlCoords c1 = make_coords<512>(gx, gy);
    const LvlCoords c2 = make_coords<256>(gx, gy);
    const LvlCoords c3 = make_coords<128>(gx, gy);

    // early prefetch (global_prefetch_b8) of the later levels' first rows so
    // their lines are in flight while level-0 gathers execute
    __builtin_prefetch(t1 + (size_t)cbase * (512u * 512u) + (size_t)c1.off0, 0, 3);
    __builtin_prefetch(t2 + (size_t)cbase * (256u * 256u) + (size_t)c2.off0, 0, 3);
    __builtin_prefetch(t3 + (size_t)cbase * (128u * 128u) + (size_t)c3.off0, 0, 3);

    float acc[8];
#pragma unroll
    for (int c = 0; c < 8; ++c) acc[c] = 0.0f;

    accum_level<1024, 8>(t0, c0, cbase, acc);
    accum_level<512, 8>(t1, c1, cbase, acc);
    accum_level<256, 8>(t2, c2, cbase, acc);
    accum_level<128, 8>(t3, c3, cbase, acc);

    // out[n][cbase+c][ho][wo]; NT stores: 128 MiB streaming output must not
    // evict the 170 MiB of L2-resident texture data.
    const int n = p >> 18;           // 512*512 = 2^18 pixels per batch image
    const int rem = p & 262143;      // ho*512 + wo
    float* o = out + (((size_t)(n * 32 + cbase)) << 18) + (size_t)rem;
#pragma unroll
    for (int c = 0; c < 8; ++c)
        __builtin_nontemporal_store(acc[c], o + ((size_t)c << 18));
}

extern "C" void kernel_launch(void* const* d_in, const int* in_sizes, int n_in,
                              void* d_out, int out_size, void* d_ws, size_t ws_size,
                              hipStream_t stream) {
    (void)in_sizes; (void)n_in; (void)out_size; (void)d_ws; (void)ws_size;
    const float* x  = (const float*)d_in[0];   // (4, 512, 512, 2)
    const float* t0 = (const float*)d_in[1];   // (32, 1024, 1024)
    const float* t1 = (const float*)d_in[2];   // (32, 512, 512)
    const float* t2 = (const float*)d_in[3];   // (32, 256, 256)
    const float* t3 = (const float*)d_in[4];   // (32, 128, 128)
    float* out = (float*)d_out;                // (4, 32, 512, 512)

    // 4*512*512 pixels / 256 threads = 4096 blocks; grid.y = 4 channel chunks
    dim3 grid(4096, 4, 1);
    dim3 block(256, 1, 1);
    grid_pyr_58128087384510_kernel<<<grid, block, 0, stream>>>(x, t0, t1, t2, t3, out);
}